// InstantNGP_31559419691686
// MI455X (gfx1250) — compile-verified
//
#include <hip/hip_runtime.h>
#include <math.h>

// ---------------------------------------------------------------------------
// InstantNGP hash-grid encoder for MI455X (gfx1250, wave32).
//
// Thread mapping: global thread g -> point = g>>4, level = g&15.
//   * 8.4M threads, ~8 outstanding global_load_b64 gathers each (deep LOADcnt
//     MLP, the real limiter for hash gathers).
//   * Whole 64MB table set is L2-resident (192MB L2) -> gathers hit L2.
//   * Output written with non-temporal hint so the 64MB stream does not
//     displace the tables from L2.
//   * Wave lanes 0..15 / 16..31 cover all levels of two points -> the two
//     float2 output stores per wave are fully coalesced 256B runs.
// ---------------------------------------------------------------------------

#define NGP_L      16
#define NGP_LOG2T  19
#define NGP_T      (1u << NGP_LOG2T)
#define NGP_MASK   (NGP_T - 1u)
#define NGP_P1     2654435761u
#define NGP_P2     805459861u

typedef float v2f __attribute__((ext_vector_type(2)));

struct LevelCells {
    float cell[NGP_L];   // (bb_max-bb_min)/res per level, computed on host in f32
};

__global__ __launch_bounds__(256, 8)
void InstantNGP_hashgrid_kernel(const float* __restrict__ xyz,
                                const v2f*  __restrict__ tables,  // [L*T] float2
                                v2f*        __restrict__ out,     // [N*16] float2 == [N, L*F]
                                LevelCells lc, int npts)
{
    // Stage the 16 points this block touches (48 floats) through LDS so the
    // 16 level-threads of each point share one coalesced global read.
    __shared__ float sxyz[48];
    const int tid = threadIdx.x;
    {
        const int idx = blockIdx.x * 48 + tid;
        if (tid < 48)
            sxyz[tid] = (idx < npts * 3) ? xyz[idx] : 0.0f;
    }
    __syncthreads();

    const int g = blockIdx.x * 256 + tid;
    if (g >= npts * 16) return;

    const int point = g >> 4;
    const int level = g & 15;
    const int lp    = tid >> 4;              // local point 0..15

    const float x = sxyz[lp * 3 + 0];
    const float y = sxyz[lp * 3 + 1];
    const float z = sxyz[lp * 3 + 2];

    // Replicate reference numerics exactly: u = xyz / cell (f32 IEEE divide),
    // cell = 1/res computed in f32 on host. floor / frac in f32.
    const float cell = lc.cell[level];
    const float ux = x / cell, uy = y / cell, uz = z / cell;
    const float fx = floorf(ux), fy = floorf(uy), fz = floorf(uz);
    const float dx = ux - fx,   dy = uy - fy,   dz = uz - fz;

    // Spatial hash with uint32 wrap; (c+1)*p == c*p + p mod 2^32.
    const unsigned hx0 = (unsigned)(int)fx;           // prime_x = 1
    const unsigned hy0 = (unsigned)(int)fy * NGP_P1;
    const unsigned hz0 = (unsigned)(int)fz * NGP_P2;
    const unsigned hx1 = hx0 + 1u;
    const unsigned hy1 = hy0 + NGP_P1;
    const unsigned hz1 = hz0 + NGP_P2;

    const float wx0 = 1.0f - dx, wy0 = 1.0f - dy, wz0 = 1.0f - dz;

    const v2f* __restrict__ tab = tables + ((size_t)level << NGP_LOG2T);

    // VERTS order: v bit2 -> x, bit1 -> y, bit0 -> z (matches reference).
    // Issue all 8 gathers, then blend; compiler keeps 8 loads in flight.
    float f0 = 0.0f, f1 = 0.0f;
#pragma unroll
    for (int v = 0; v < 8; ++v) {
        const unsigned hx = (v & 4) ? hx1 : hx0;
        const unsigned hy = (v & 2) ? hy1 : hy0;
        const unsigned hz = (v & 1) ? hz1 : hz0;
        const unsigned h  = (hx ^ hy ^ hz) & NGP_MASK;
        const v2f e = tab[h];                      // global_load_b64 gather (L2-resident)
        const float wx = (v & 4) ? dx : wx0;
        const float wy = (v & 2) ? dy : wy0;
        const float wz = (v & 1) ? dz : wz0;
        const float w  = (wx * wy) * wz;           // reference prod order x,y,z
        f0 = fmaf(w, e.x, f0);
        f1 = fmaf(w, e.y, f1);
    }

    // out[point, level*2 + {0,1}] : float2 index = point*16 + level.
    // Non-temporal store: 64MB write stream must not evict tables from L2.
    v2f r; r.x = f0; r.y = f1;
    __builtin_nontemporal_store(r, out + ((size_t)point << 4) + level);
}

extern "C" void kernel_launch(void* const* d_in, const int* in_sizes, int n_in,
                              void* d_out, int out_size, void* d_ws, size_t ws_size,
                              hipStream_t stream)
{
    const float* xyz    = (const float*)d_in[0];
    const v2f*   tables = (const v2f*)  d_in[1];
    v2f*         out    = (v2f*)        d_out;

    const int npts = in_sizes[0] / 3;

    // Replicate reference resolution chain in double precision (same libm
    // path CPython uses), truncate to f32, then cell = 1.0f / res in f32 —
    // bit-identical to np.float32 RES and cell arrays in the reference.
    LevelCells lc;
    const double B = exp((log(2048.0) - log(16.0)) / 15.0);
    for (int i = 0; i < NGP_L; ++i) {
        const double res = floor(16.0 * pow(B, (double)i));
        lc.cell[i] = 1.0f / (float)res;
    }

    const int nthreads = npts * 16;
    const int blocks   = (nthreads + 255) / 256;
    hipLaunchKernelGGL(InstantNGP_hashgrid_kernel,
                       dim3(blocks), dim3(256), 0, stream,
                       xyz, tables, out, lc, npts);
}